// SE2MultiSelfAttention_39651138077216
// MI455X (gfx1250) — compile-verified
//
#include <hip/hip_runtime.h>
#include <math.h>

// Problem constants (match reference): B, MAXM, DIM, NH, QD, H, W
#define Bq   8
#define Mq   4
#define DIMq 128
#define NHq  8
#define QDq  16
#define Nq   256   // H*W = 16*16

typedef __attribute__((ext_vector_type(2))) float v2f;
typedef __attribute__((ext_vector_type(8))) float v8f;

// D = A(16x4, f32) * B(4x16, f32) + C(16x16, f32)  -- CDNA5 V_WMMA_F32_16X16X4_F32
// NEG on A/B is illegal for F32 WMMA (only C-neg), so pass false and handle
// complex signs by combining separate accumulators.
__device__ __forceinline__ v8f wmma4(v2f a, v2f b, v8f c) {
    return __builtin_amdgcn_wmma_f32_16x16x4_f32(false, a, false, b, (short)0, c,
                                                 false, false);
}

// -------------------------------------------------------------------------
// Kernel 1: E[t,b,m,h] (QDxN) = emb[t,m,h] (QDxDIM) @ x[b,m] (DIMxN), complex.
// grid.x encodes ((t*B+b)*M+m)*NH+h, grid.y = j-tile (16 of them). 1 wave/block.
// -------------------------------------------------------------------------
__global__ void __launch_bounds__(32)
k_embed(const float* __restrict__ xr, const float* __restrict__ xi,
        const float* __restrict__ er, const float* __restrict__ ei,
        float* __restrict__ Er, float* __restrict__ Ei) {
    int idx = blockIdx.x;                 // ((t*B+b)*M+m)*NH+h
    int h = idx % NHq;
    int m = (idx / NHq) % Mq;
    int t = idx / (NHq * Mq * Bq);
    int b = (idx / (NHq * Mq)) % Bq;
    int j0 = blockIdx.y * 16;
    int lane = threadIdx.x;
    int r  = lane & 15;                   // A row / B col / C col
    int kh = (lane >> 4) << 1;            // K offset 0 or 2
    int ch = (lane >> 4) * 8;             // C row offset 0 or 8

    const float* ar_b = er + ((size_t)((t * Mq + m) * NHq + h)) * QDq * DIMq;
    const float* ai_b = ei + ((size_t)((t * Mq + m) * NHq + h)) * QDq * DIMq;
    const float* br_b = xr + ((size_t)(b * Mq + m)) * DIMq * Nq;
    const float* bi_b = xi + ((size_t)(b * Mq + m)) * DIMq * Nq;

    v8f a0 = {}, a1 = {}, a2 = {}, a3 = {};
    for (int k0 = 0; k0 < DIMq; k0 += 4) {
        int k = k0 + kh;
        v2f ar, ai, br, bi;
        ar.x = ar_b[r * DIMq + k];     ar.y = ar_b[r * DIMq + k + 1];
        ai.x = ai_b[r * DIMq + k];     ai.y = ai_b[r * DIMq + k + 1];
        br.x = br_b[k * Nq + j0 + r];  br.y = br_b[(k + 1) * Nq + j0 + r];
        bi.x = bi_b[k * Nq + j0 + r];  bi.y = bi_b[(k + 1) * Nq + j0 + r];
        a0 = wmma4(ar, br, a0);   // Re*Re
        a1 = wmma4(ai, bi, a1);   // Im*Im
        a2 = wmma4(ar, bi, a2);   // Re*Im
        a3 = wmma4(ai, br, a3);   // Im*Re
    }
    float* Ero = Er + (size_t)idx * QDq * Nq;
    float* Eio = Ei + (size_t)idx * QDq * Nq;
    for (int v = 0; v < 8; ++v) {
        int q = ch + v;
        Ero[q * Nq + j0 + r] = a0[v] - a1[v];
        Eio[q * Nq + j0 + r] = a2[v] + a3[v];
    }
}

// -------------------------------------------------------------------------
// Kernel 2: per (b,h,i-tile): A[i,j] = sum_m ( conj(E0)^T K + QE[m,i]*pe[m,i,j] ),
// then Aw = softmax(|A|/4) over j, and PA[b,m,h,i] = sum_j pe[m,i,j]*Aw[i,j].
// 512 threads = 16 waves; wave w owns j-tile w for the GEMM, row w for softmax.
// -------------------------------------------------------------------------
__global__ void __launch_bounds__(512)
k_attn(const float* __restrict__ Er, const float* __restrict__ Ei,
       const float* __restrict__ encr, const float* __restrict__ enci,
       float* __restrict__ Aw, float* __restrict__ PAr, float* __restrict__ PAi) {
    __shared__ float sQEr[Mq][16];
    __shared__ float sQEi[Mq][16];
    __shared__ float sAbs[16][Nq];

    int bid = blockIdx.x;
    int it = bid & 15;
    int h  = (bid >> 4) & (NHq - 1);
    int b  = bid >> 7;
    int i0 = it * 16;
    int tid = threadIdx.x;

    // ---- Phase 1: QE[m, i_local] = sum_q conj(E0[q,i]) * enc0[m,h,q]
    if (tid < Mq * 16) {
        int m  = tid >> 4;
        int il = tid & 15;
        const float* e0r = Er + ((size_t)(((0 * Bq + b) * Mq + m) * NHq + h)) * QDq * Nq;
        const float* e0i = Ei + ((size_t)(((0 * Bq + b) * Mq + m) * NHq + h)) * QDq * Nq;
        const float* c0r = encr + ((size_t)((0 * Mq + m) * NHq + h)) * QDq;
        const float* c0i = enci + ((size_t)((0 * Mq + m) * NHq + h)) * QDq;
        float qr = 0.f, qi = 0.f;
        for (int q = 0; q < QDq; ++q) {
            float e_r = e0r[q * Nq + i0 + il];
            float e_i = e0i[q * Nq + i0 + il];
            float c_r = c0r[q], c_i = c0i[q];
            qr += e_r * c_r + e_i * c_i;   // Re( conj(E0)*enc )
            qi += e_r * c_i - e_i * c_r;   // Im( conj(E0)*enc )
        }
        sQEr[m][il] = qr;
        sQEi[m][il] = qi;
    }
    __syncthreads();

    // ---- Phase 2: WMMA A = conj(E0)^T K, accumulate over m, inject pe*QE term
    int wave = tid >> 5;
    int lane = tid & 31;
    int j0 = wave * 16;
    int r  = lane & 15;
    int kh = (lane >> 4) << 1;
    int ch = (lane >> 4) * 8;

    v8f p1 = {}, p2 = {}, p3 = {}, p4 = {};
    for (int m = 0; m < Mq; ++m) {
        const float* e0r = Er + ((size_t)(((0 * Bq + b) * Mq + m) * NHq + h)) * QDq * Nq;
        const float* e0i = Ei + ((size_t)(((0 * Bq + b) * Mq + m) * NHq + h)) * QDq * Nq;
        const float* kr  = Er + ((size_t)(((1 * Bq + b) * Mq + m) * NHq + h)) * QDq * Nq;
        const float* ki  = Ei + ((size_t)(((1 * Bq + b) * Mq + m) * NHq + h)) * QDq * Nq;
        for (int k0 = 0; k0 < QDq; k0 += 4) {
            int q = k0 + kh;
            v2f ar, ai, br, bi;
            // A[i,q] = E0[q, i]  (transposed read; conj handled via accum combine)
            ar.x = e0r[q * Nq + i0 + r];       ar.y = e0r[(q + 1) * Nq + i0 + r];
            ai.x = e0i[q * Nq + i0 + r];       ai.y = e0i[(q + 1) * Nq + i0 + r];
            br.x = kr[q * Nq + j0 + r];        br.y = kr[(q + 1) * Nq + j0 + r];
            bi.x = ki[q * Nq + j0 + r];        bi.y = ki[(q + 1) * Nq + j0 + r];
            p1 = wmma4(ar, br, p1);   // E0r.Kr
            p2 = wmma4(ai, bi, p2);   // E0i.Ki
            p3 = wmma4(ar, bi, p3);   // E0r.Ki
            p4 = wmma4(ai, br, p4);   // E0i.Kr
        }
        // A += QE[m,i] * exp(i*m*theta(i,j))  (add Re into p1, Im into p3)
        for (int v = 0; v < 8; ++v) {
            int il = ch + v;
            int i = i0 + il;
            int j = j0 + r;
            float dy = (float)(j >> 4) - (float)(i >> 4);
            float dx = (float)(j & 15) - (float)(i & 15);
            float th = atan2f(dy, dx);
            float s, c;
            sincosf((float)m * th, &s, &c);
            float qr = sQEr[m][il], qi = sQEi[m][il];
            p1[v] += qr * c - qi * s;
            p3[v] += qr * s + qi * c;
        }
    }
    for (int v = 0; v < 8; ++v) {
        float re = p1[v] + p2[v];      // A_re = E0r.Kr + E0i.Ki  (conj)
        float im = p3[v] - p4[v];      // A_im = E0r.Ki - E0i.Kr
        sAbs[ch + v][j0 + r] = sqrtf(re * re + im * im) * 0.25f;  // |A|/sqrt(QD)
    }
    __syncthreads();

    // ---- Phase 3: per-row softmax + PA reduction. Wave w -> local row w.
    {
        int row = wave;
        int i = i0 + row;
        const float* rowp = sAbs[row];
        float mx = -3.402823466e38f;
        for (int j = lane; j < Nq; j += 32) mx = fmaxf(mx, rowp[j]);
        for (int d = 16; d >= 1; d >>= 1) mx = fmaxf(mx, __shfl_xor(mx, d, 32));
        float sum = 0.f;
        for (int j = lane; j < Nq; j += 32) sum += expf(rowp[j] - mx);
        for (int d = 16; d >= 1; d >>= 1) sum += __shfl_xor(sum, d, 32);
        float inv = 1.f / sum;

        float pa0 = 0.f;
        float pr1 = 0.f, pi1 = 0.f, pr2 = 0.f, pi2 = 0.f, pr3 = 0.f, pi3 = 0.f;
        float* awp = Aw + ((size_t)(b * NHq + h) * Nq + i) * Nq;
        for (int j = lane; j < Nq; j += 32) {
            float aw = expf(rowp[j] - mx) * inv;
            awp[j] = aw;
            float dy = (float)(j >> 4) - (float)(i >> 4);
            float dx = (float)(j & 15) - (float)(i & 15);
            float th = atan2f(dy, dx);
            float s1, c1;
            sincosf(th, &s1, &c1);
            float c2 = c1 * c1 - s1 * s1, s2 = 2.f * s1 * c1;       // angle 2*th
            float c3 = c2 * c1 - s2 * s1, s3 = s2 * c1 + c2 * s1;   // angle 3*th
            pa0 += aw;
            pr1 += aw * c1;  pi1 += aw * s1;
            pr2 += aw * c2;  pi2 += aw * s2;
            pr3 += aw * c3;  pi3 += aw * s3;
        }
        for (int d = 16; d >= 1; d >>= 1) {
            pa0 += __shfl_xor(pa0, d, 32);
            pr1 += __shfl_xor(pr1, d, 32);  pi1 += __shfl_xor(pi1, d, 32);
            pr2 += __shfl_xor(pr2, d, 32);  pi2 += __shfl_xor(pi2, d, 32);
            pr3 += __shfl_xor(pr3, d, 32);  pi3 += __shfl_xor(pi3, d, 32);
        }
        if (lane == 0) {
            size_t s0 = ((size_t)(b * Mq + 0) * NHq + h) * Nq + i;
            size_t s1i = ((size_t)(b * Mq + 1) * NHq + h) * Nq + i;
            size_t s2i = ((size_t)(b * Mq + 2) * NHq + h) * Nq + i;
            size_t s3i = ((size_t)(b * Mq + 3) * NHq + h) * Nq + i;
            PAr[s0]  = pa0;  PAi[s0]  = 0.f;
            PAr[s1i] = pr1;  PAi[s1i] = pi1;
            PAr[s2i] = pr2;  PAi[s2i] = pi2;
            PAr[s3i] = pr3;  PAi[s3i] = pi3;
        }
    }
}

// -------------------------------------------------------------------------
// Kernel 3: res[b,m,h,q,i] = sum_j V[q,j]*Aw[i,j] + enc1[m,h,q]*PA[b,m,h,i]
// stored as res[b][m][h*16+q][i]. grid.x = (b*M+m)*NH+h, grid.y = i-tile.
// -------------------------------------------------------------------------
__global__ void __launch_bounds__(32)
k_res(const float* __restrict__ Er, const float* __restrict__ Ei,
      const float* __restrict__ Aw,
      const float* __restrict__ PAr, const float* __restrict__ PAi,
      const float* __restrict__ encr, const float* __restrict__ enci,
      float* __restrict__ resr, float* __restrict__ resi) {
    int idx = blockIdx.x;                  // (b*M+m)*NH+h
    int h = idx % NHq;
    int m = (idx / NHq) % Mq;
    int b = idx / (NHq * Mq);
    int i0 = blockIdx.y * 16;
    int lane = threadIdx.x;
    int r  = lane & 15;
    int kh = (lane >> 4) << 1;
    int ch = (lane >> 4) * 8;

    const float* vr  = Er + ((size_t)(((2 * Bq + b) * Mq + m) * NHq + h)) * QDq * Nq;
    const float* vi  = Ei + ((size_t)(((2 * Bq + b) * Mq + m) * NHq + h)) * QDq * Nq;
    const float* awb = Aw + ((size_t)(b * NHq + h)) * Nq * Nq;

    v8f cr = {}, ci = {};
    for (int k0 = 0; k0 < Nq; k0 += 4) {
        int j = k0 + kh;
        v2f ar, ai, bb;
        ar.x = vr[r * Nq + j];              ar.y = vr[r * Nq + j + 1];
        ai.x = vi[r * Nq + j];              ai.y = vi[r * Nq + j + 1];
        bb.x = awb[(i0 + r) * Nq + j];      bb.y = awb[(i0 + r) * Nq + j + 1];
        cr = wmma4(ar, bb, cr);
        ci = wmma4(ai, bb, ci);
    }
    const float* e1r = encr + ((size_t)((1 * Mq + m) * NHq + h)) * QDq;
    const float* e1i = enci + ((size_t)((1 * Mq + m) * NHq + h)) * QDq;
    const float* pr  = PAr + ((size_t)(b * Mq + m) * NHq + h) * Nq;
    const float* pi  = PAi + ((size_t)(b * Mq + m) * NHq + h) * Nq;
    float* ror = resr + ((size_t)(b * Mq + m) * (NHq * QDq) + h * QDq) * Nq;
    float* roi = resi + ((size_t)(b * Mq + m) * (NHq * QDq) + h * QDq) * Nq;
    int i = i0 + r;
    float par = pr[i], pai = pi[i];
    for (int v = 0; v < 8; ++v) {
        int q = ch + v;
        float er_ = e1r[q], ei_ = e1i[q];
        ror[q * Nq + i] = cr[v] + er_ * par - ei_ * pai;
        roi[q * Nq + i] = ci[v] + er_ * pai + ei_ * par;
    }
}

// -------------------------------------------------------------------------
// Kernel 4: out[b,m] (DIMxN) = w_out[m] (DIMx128) @ res[b,m] (128xN), complex.
// grid = (b*M+m, Dtile(8), ntile(16)); 1 wave/block. Writes real+imag planes.
// -------------------------------------------------------------------------
__global__ void __launch_bounds__(32)
k_out(const float* __restrict__ wr, const float* __restrict__ wi,
      const float* __restrict__ resr, const float* __restrict__ resi,
      float* __restrict__ out) {
    const int K = NHq * QDq;               // 128
    int bm = blockIdx.x;
    int m = bm % Mq;
    int d0 = blockIdx.y * 16;
    int n0 = blockIdx.z * 16;
    int lane = threadIdx.x;
    int r  = lane & 15;
    int kh = (lane >> 4) << 1;
    int ch = (lane >> 4) * 8;

    const float* war = wr + (size_t)m * DIMq * K;
    const float* wai = wi + (size_t)m * DIMq * K;
    const float* rbr = resr + (size_t)bm * K * Nq;
    const float* rbi = resi + (size_t)bm * K * Nq;

    v8f a0 = {}, a1 = {}, a2 = {}, a3 = {};
    for (int k0 = 0; k0 < K; k0 += 4) {
        int k = k0 + kh;
        v2f ar, ai, br, bi;
        ar.x = war[(d0 + r) * K + k];   ar.y = war[(d0 + r) * K + k + 1];
        ai.x = wai[(d0 + r) * K + k];   ai.y = wai[(d0 + r) * K + k + 1];
        br.x = rbr[k * Nq + n0 + r];    br.y = rbr[(k + 1) * Nq + n0 + r];
        bi.x = rbi[k * Nq + n0 + r];    bi.y = rbi[(k + 1) * Nq + n0 + r];
        a0 = wmma4(ar, br, a0);
        a1 = wmma4(ai, bi, a1);
        a2 = wmma4(ar, bi, a2);
        a3 = wmma4(ai, br, a3);
    }
    size_t plane = (size_t)Bq * Mq * DIMq * Nq;
    float* o_re = out + (size_t)bm * DIMq * Nq;
    float* o_im = o_re + plane;
    for (int v = 0; v < 8; ++v) {
        int d = d0 + ch + v;
        o_re[(size_t)d * Nq + n0 + r] = a0[v] - a1[v];
        o_im[(size_t)d * Nq + n0 + r] = a2[v] + a3[v];
    }
}

// -------------------------------------------------------------------------
extern "C" void kernel_launch(void* const* d_in, const int* in_sizes, int n_in,
                              void* d_out, int out_size, void* d_ws, size_t ws_size,
                              hipStream_t stream) {
    const float* xr   = (const float*)d_in[0];
    const float* xi   = (const float*)d_in[1];
    const float* embr = (const float*)d_in[2];
    const float* embi = (const float*)d_in[3];
    const float* encr = (const float*)d_in[4];
    const float* enci = (const float*)d_in[5];
    const float* outr = (const float*)d_in[6];
    const float* outi = (const float*)d_in[7];

    float* ws = (float*)d_ws;
    const size_t EN   = (size_t)3 * Bq * Mq * NHq * QDq * Nq;   // 3,145,728
    const size_t AWN  = (size_t)Bq * NHq * Nq * Nq;             // 4,194,304
    const size_t PAN  = (size_t)Bq * Mq * NHq * Nq;             //    65,536
    const size_t RESN = (size_t)Bq * Mq * NHq * QDq * Nq;       // 1,048,576
    float* Er   = ws;
    float* Ei   = Er + EN;
    float* Aw   = Ei + EN;
    float* PAr  = Aw + AWN;
    float* PAi  = PAr + PAN;
    float* resr = PAi + PAN;
    float* resi = resr + RESN;

    k_embed<<<dim3(3 * Bq * Mq * NHq, 16), 32, 0, stream>>>(xr, xi, embr, embi, Er, Ei);
    k_attn<<<dim3(Bq * NHq * 16), 512, 0, stream>>>(Er, Ei, encr, enci, Aw, PAr, PAi);
    k_res<<<dim3(Bq * Mq * NHq, 16), 32, 0, stream>>>(Er, Ei, Aw, PAr, PAi, encr, enci,
                                                      resr, resi);
    k_out<<<dim3(Bq * Mq, 8, 16), 32, 0, stream>>>(outr, outi, resr, resi, (float*)d_out);
}